// HyLoRADALinear_14869176779243
// MI455X (gfx1250) — compile-verified
//
#include <hip/hip_runtime.h>
#include <math.h>

typedef __attribute__((ext_vector_type(2))) float v2f;
typedef __attribute__((ext_vector_type(8))) float v8f;

#define D_IN   4096
#define D_OUT  4096
#define RANKV  8
#define NROWS  8192            // B*S = 4*2048
#define SCALE  2.0f            // alpha/rank = 16/8
#define EPSV   1e-8f

// ---------------------------------------------------------------------------
// Kernel 1: DoRA column scale.  One block per output row o.
//   mag_scale[o] = magnitude[o] / (||W[o,:] + s * B[o,:] @ A||_2 + eps)
// Streams base_weight (64 MiB) coalesced; lora_A/lora_B are cache-resident.
// ---------------------------------------------------------------------------
__global__ __launch_bounds__(256) void dora_magscale_kernel(
    const float* __restrict__ base_weight,
    const float* __restrict__ lora_A,
    const float* __restrict__ lora_B,
    const float* __restrict__ magnitude,
    float* __restrict__ mag_scale)
{
    const int o   = blockIdx.x;
    const int tid = threadIdx.x;

    float bv[RANKV];
#pragma unroll
    for (int r = 0; r < RANKV; ++r) bv[r] = lora_B[o * RANKV + r];

    const float4* wrow = (const float4*)(base_weight + (size_t)o * D_IN);
    float acc = 0.0f;

    for (int i4 = tid; i4 < D_IN / 4; i4 += 256) {
        float4 w = wrow[i4];
        float dx = 0.f, dy = 0.f, dz = 0.f, dw = 0.f;
#pragma unroll
        for (int r = 0; r < RANKV; ++r) {
            float4 a = ((const float4*)(lora_A + (size_t)r * D_IN))[i4];
            dx += bv[r] * a.x; dy += bv[r] * a.y;
            dz += bv[r] * a.z; dw += bv[r] * a.w;
        }
        float ux = w.x + SCALE * dx;
        float uy = w.y + SCALE * dy;
        float uz = w.z + SCALE * dz;
        float uw = w.w + SCALE * dw;
        acc += ux * ux + uy * uy + uz * uz + uw * uw;
    }

    __shared__ float red[256];
    red[tid] = acc;
    __syncthreads();
#pragma unroll
    for (int s = 128; s > 0; s >>= 1) {
        if (tid < s) red[tid] += red[tid + s];
        __syncthreads();
    }
    if (tid == 0)
        mag_scale[o] = magnitude[o] / (sqrtf(red[0]) + EPSV);
}

// ---------------------------------------------------------------------------
// Kernel 2: fused LoRA path.  One block per 16 (b,s) rows; 8 waves.
//   Phase A: lora_x[16][8] = x_tile @ A^T  (coalesced float4 streaming,
//            wave32 shfl_xor butterfly reduction, scaled by s, into LDS)
//   Phase B: per-wave 16x16 output tiles:
//            delta = lora_x(16x8) @ B^T(8x16)  via 2x V_WMMA_F32_16X16X4_F32
//            out = (base_output + delta) * mag_scale
// ---------------------------------------------------------------------------
__global__ __launch_bounds__(256) void dora_fused_kernel(
    const float* __restrict__ x,
    const float* __restrict__ base_output,
    const float* __restrict__ lora_A,
    const float* __restrict__ lora_B,
    const float* __restrict__ mag_scale,
    float* __restrict__ out)
{
    const int row0 = blockIdx.x * 16;
    const int tid  = threadIdx.x;
    const int lane = tid & 31;
    const int wave = tid >> 5;          // 0..7

    __shared__ float lx[16][RANKV];     // scaled lora_x tile

    // ---- Phase A: two rows per wave ----
#pragma unroll
    for (int rr = 0; rr < 2; ++rr) {
        const int m = wave * 2 + rr;    // local row 0..15
        const float4* xr = (const float4*)(x + (size_t)(row0 + m) * D_IN);
        float p[RANKV];
#pragma unroll
        for (int r = 0; r < RANKV; ++r) p[r] = 0.0f;

        for (int i4 = lane; i4 < D_IN / 4; i4 += 32) {
            float4 xv = xr[i4];
#pragma unroll
            for (int r = 0; r < RANKV; ++r) {
                float4 a = ((const float4*)(lora_A + (size_t)r * D_IN))[i4];
                p[r] += xv.x * a.x + xv.y * a.y + xv.z * a.z + xv.w * a.w;
            }
        }
#pragma unroll
        for (int off = 16; off > 0; off >>= 1) {
#pragma unroll
            for (int r = 0; r < RANKV; ++r)
                p[r] += __shfl_xor(p[r], off, 32);
        }
        if (lane < RANKV) lx[m][lane] = SCALE * p[lane];
    }
    __syncthreads();

    // ---- Phase B: WMMA expand + fuse ----
    // A-matrix 16x4 f32 layout: lanes 0-15 -> M=lane, K={k0,k0+1} with k0=0;
    //                           lanes 16-31 -> M=lane-16, K={2,3}.
    const int am = lane & 15;
    const int ak = (lane >> 4) * 2;
    v2f a0, a1;
    a0.x = lx[am][ak + 0]; a0.y = lx[am][ak + 1];   // K = 0..3
    a1.x = lx[am][ak + 4]; a1.y = lx[am][ak + 5];   // K = 4..7

    // B-matrix 4x16 f32 layout: lanes 0-15 -> N=lane, K={0,1};
    //                           lanes 16-31 -> N=lane-16, K={2,3}.
    const int bn   = lane & 15;
    const int bk   = (lane >> 4) * 2;
    const int mrow = (lane >> 4) * 8;   // C/D: lanes 16-31 hold M=j+8

    for (int n0 = wave * 16; n0 < D_OUT; n0 += 8 * 16) {
        const float* bp = lora_B + (size_t)(n0 + bn) * RANKV + bk;
        v2f b0, b1;
        b0.x = bp[0]; b0.y = bp[1];     // K = 0..3
        b1.x = bp[4]; b1.y = bp[5];     // K = 4..7

        v8f c = {};
        c = __builtin_amdgcn_wmma_f32_16x16x4_f32(false, a0, false, b0,
                                                  (short)0, c, false, false);
        c = __builtin_amdgcn_wmma_f32_16x16x4_f32(false, a1, false, b1,
                                                  (short)0, c, false, false);

        const int   ncol = n0 + (lane & 15);
        const float mg   = mag_scale[ncol];
#pragma unroll
        for (int j = 0; j < 8; ++j) {
            const size_t idx = (size_t)(row0 + mrow + j) * D_OUT + ncol;
            out[idx] = (base_output[idx] + c[j]) * mg;
        }
    }
}

// ---------------------------------------------------------------------------
// kernel_launch
//   d_in order: x, base_output, base_weight, lora_A, lora_B, magnitude
//   d_ws: first 4096 floats hold mag_scale
// ---------------------------------------------------------------------------
extern "C" void kernel_launch(void* const* d_in, const int* in_sizes, int n_in,
                              void* d_out, int out_size, void* d_ws, size_t ws_size,
                              hipStream_t stream) {
    (void)in_sizes; (void)n_in; (void)out_size; (void)ws_size;
    const float* x           = (const float*)d_in[0];
    const float* base_output = (const float*)d_in[1];
    const float* base_weight = (const float*)d_in[2];
    const float* lora_A      = (const float*)d_in[3];
    const float* lora_B      = (const float*)d_in[4];
    const float* magnitude   = (const float*)d_in[5];
    float*       out         = (float*)d_out;
    float*       mag_scale   = (float*)d_ws;

    dora_magscale_kernel<<<D_OUT, 256, 0, stream>>>(
        base_weight, lora_A, lora_B, magnitude, mag_scale);

    dora_fused_kernel<<<NROWS / 16, 256, 0, stream>>>(
        x, base_output, lora_A, lora_B, mag_scale, out);
}